// StructuralInteractions_2843268350403
// MI455X (gfx1250) — compile-verified
//
#include <hip/hip_runtime.h>
#include <hip/hip_bf16.h>

typedef __attribute__((ext_vector_type(2))) float v2f;
typedef __attribute__((ext_vector_type(4))) float v4f;
typedef __attribute__((ext_vector_type(8))) float v8f;

#define NG      64
#define NPG     512
#define KNN     17                     // K+1 (self loop)
#define E_TOTAL (NG * NPG * KNN)       // 557056 edges
#define RBF_N   128
#define RADIUS_F 8.0f
#define SSTRIDE 516                    // strip row stride (floats): conflict-free + 16B-aligned rows

// ---------------------------------------------------------------------------
// Kernel A: one workgroup = (graph g, 16-row strip). 512 threads = 16 waves.
// Phase 0: per-column |p|^2 into LDS.
// Phase 1: each wave computes two 16x16 tiles of p_i . p_j via
//          V_WMMA_F32_16X16X4_F32 (K=3 padded to 4), converts to squared
//          distance with the LDS norm table, stores a 16x512 strip in LDS.
// Phase 2: threads 0..15 each run a b128-vectorized top-17 insertion over one
//          row, emit edge_index (src/dst, NT stores) and sqrt-distances (ws).
// ---------------------------------------------------------------------------
__global__ __launch_bounds__(512)
void knn_dist_kernel(const float* __restrict__ pos,
                     float* __restrict__ out,       // d_out base (edge_index region)
                     float* __restrict__ dist_ws) { // E_TOTAL floats
    __shared__ float cnorm[NPG];
    __shared__ float strip[16 * SSTRIDE];

    const int g        = blockIdx.x >> 5;     // graph
    const int strip_id = blockIdx.x & 31;     // 32 strips of 16 rows
    const int row_base = strip_id * 16;
    const int t    = threadIdx.x;
    const int lane = t & 31;
    const int wave = t >> 5;
    const float* gpos = pos + (size_t)g * NPG * 3;

    // Phase 0: column norms (512 threads -> 512 columns)
    {
        const float x = gpos[t * 3 + 0];
        const float y = gpos[t * 3 + 1];
        const float z = gpos[t * 3 + 2];
        cnorm[t] = x * x + y * y + z * z;
    }
    __syncthreads();

    // A fragment: 16x4 fp32. Lanes 0-15: M=lane, K={0,1}=(x,y); lanes 16-31: K={2,3}=(z,0).
    const int arow = row_base + (lane & 15);
    const float ax = gpos[arow * 3 + 0];
    const float ay = gpos[arow * 3 + 1];
    const float az = gpos[arow * 3 + 2];
    v2f a;
    a.x = (lane < 16) ? ax : az;
    a.y = (lane < 16) ? ay : 0.0f;

    // Phase 1: each wave owns column tiles {wave, wave+16}
#pragma unroll
    for (int tt = 0; tt < 2; ++tt) {
        const int col_base = (wave + tt * 16) * 16;
        const int bcol = col_base + (lane & 15);
        const float bx = gpos[bcol * 3 + 0];
        const float by = gpos[bcol * 3 + 1];
        const float bz = gpos[bcol * 3 + 2];
        v2f b;
        b.x = (lane < 16) ? bx : bz;
        b.y = (lane < 16) ? by : 0.0f;

        v8f acc = {};
        acc = __builtin_amdgcn_wmma_f32_16x16x4_f32(
            /*neg_a=*/false, a, /*neg_b=*/false, b,
            /*c_mod=*/(short)0, acc, /*reuse_a=*/false, /*reuse_b=*/false);

        // C/D layout: VGPR v, lane l -> M = v + 8*(l>=16), N = l%16
        const int hl = lane >> 4;
        const int n  = lane & 15;
#pragma unroll
        for (int v = 0; v < 8; ++v) {
            const int m = v + 8 * hl;
            const float d2 = cnorm[row_base + m] + cnorm[col_base + n] - 2.0f * acc[v];
            strip[m * SSTRIDE + col_base + n] = d2;
        }
    }
    __syncthreads();

    // Phase 2: top-17 smallest per row (ascending, stable on ties -> matches top_k(-d2))
    if (t < 16) {
        const int row = row_base + t;
        float bd[KNN];
        int   bi[KNN];
#pragma unroll
        for (int k = 0; k < KNN; ++k) { bd[k] = 3.0e38f; bi[k] = 0; }

        // Row base is 16B-aligned (SSTRIDE*4 = 2064 bytes): scan with ds_load_b128.
        const v4f* srow4 = (const v4f*)&strip[t * SSTRIDE];
        for (int j4 = 0; j4 < NPG / 4; ++j4) {
            const v4f dv = srow4[j4];
#pragma unroll
            for (int c = 0; c < 4; ++c) {
                const float d = dv[c];
                if (d < bd[KNN - 1]) {
                    float cd = d; int ci = j4 * 4 + c;
#pragma unroll
                    for (int k = 0; k < KNN; ++k) {
                        if (cd < bd[k]) {
                            const float td = bd[k]; const int ti = bi[k];
                            bd[k] = cd; bi[k] = ci;
                            cd = td;   ci = ti;
                        }
                    }
                }
            }
        }

        const int ebase = (g * NPG + row) * KNN;
#pragma unroll
        for (int k = 0; k < KNN; ++k) {
            // edge_index: streamed once, never re-read -> non-temporal
            __builtin_nontemporal_store((float)(bi[k] + g * NPG), &out[ebase + k]);
            __builtin_nontemporal_store((float)(row   + g * NPG), &out[E_TOTAL + ebase + k]);
            // dist_ws IS re-read by kernel B -> regular store (keep in L2, 2.2 MB << 192 MB)
            dist_ws[ebase + k] = sqrtf(fmaxf(bd[k], 1e-12f));
        }
    }
}

// ---------------------------------------------------------------------------
// Kernel B: streaming RBF embedding + mask. 256 threads = 8 edges/block,
// each 32-lane group covers one edge's 128 channels as one NT b128 store.
// Pure bandwidth kernel: ~285 MB of write-once stores (NT: don't churn L2),
// exp via v_exp_f32; dist_ws reads hit L2 from kernel A.
// ---------------------------------------------------------------------------
__global__ __launch_bounds__(256)
void rbf_kernel(const float* __restrict__ dist_ws,
                const float* __restrict__ bias,
                float* __restrict__ out) {
    const int t    = threadIdx.x;
    const int edge = blockIdx.x * 8 + (t >> 5);
    const int q    = t & 31;                    // channel group: channels 4q..4q+3

    const float dist = dist_ws[edge];
    const bool  msk  = dist <= RADIUS_F;

    const float step  = RADIUS_F / (float)(RBF_N - 1);
    const float coeff = -0.5f / (step * step);

    v4f r;
#pragma unroll
    for (int c = 0; c < 4; ++c) {
        const int ch = q * 4 + c;
        const float diff = dist - step * (float)ch;
        const float v = __expf(coeff * diff * diff) + bias[ch];
        r[c] = msk ? v : 0.0f;
    }

    float* dst = out + (size_t)2 * E_TOTAL + (size_t)edge * RBF_N + q * 4;
    __builtin_nontemporal_store(r, (v4f*)dst);

    if (q == 0) {
        __builtin_nontemporal_store(msk ? 1.0f : 0.0f,
            &out[(size_t)2 * E_TOTAL + (size_t)E_TOTAL * RBF_N + edge]);
    }
}

// ---------------------------------------------------------------------------
extern "C" void kernel_launch(void* const* d_in, const int* in_sizes, int n_in,
                              void* d_out, int out_size, void* d_ws, size_t ws_size,
                              hipStream_t stream) {
    const float* pos  = (const float*)d_in[0];   // [NG*NPG, 3] fp32
    const float* bias = (const float*)d_in[1];   // [128] fp32
    // d_in[2] = num_graphs (64), fixed by the reference problem size.
    float* out     = (float*)d_out;
    float* dist_ws = (float*)d_ws;               // needs E_TOTAL*4 = 2.23 MB

    knn_dist_kernel<<<NG * 32, 512, 0, stream>>>(pos, out, dist_ws);
    rbf_kernel<<<E_TOTAL / 8, 256, 0, stream>>>(dist_ws, bias, out);
}